// SliceNApply_36962488550094
// MI455X (gfx1250) — compile-verified
//
#include <hip/hip_runtime.h>
#include <stdint.h>

// ---------------------------------------------------------------------------
// Bilateral-grid slice + apply (HDRNet) for MI455X / gfx1250.
// Memory-bound: 112 MB streamed @ 23.3 TB/s => ~4.8us floor. Grid (96 KB per
// batch) is staged into LDS via gfx1250 async global->LDS DMA, transposed to
// [z][y][x][c] with a 16B pad per z-slice (z-stride 3076 floats, != 0 mod 64
// banks) so the guide-dependent z gathers don't serialize on LDS banks.
// Streaming tensors use non-temporal loads/stores so the caches stay
// dedicated to the grid.
// ---------------------------------------------------------------------------

#define TPB                 256
#define BLOCKS_PER_BATCH    256
#define GD                  8
#define GH                  16
#define GW                  16
#define GC                  12
#define GRID_PER_BATCH      (GC * GD * GH * GW)   // 24576 floats = 96 KB
#define ZSTRIDE             (GH * GW * GC + 4)    // 3076 floats (pad: 4%64 bank skew)
#define LDS_FLOATS          (GD * ZSTRIDE)        // 24608 floats = 98432 B
#define IMG_H               1024
#define IMG_W               1024
#define IMG_HW              (IMG_H * IMG_W)
#define PIX_PER_BLOCK       (IMG_HW / BLOCKS_PER_BATCH)   // 4096
#define SWEEPS              (PIX_PER_BLOCK / (TPB * 4))   // 4

// Native clang vector (required by __builtin_nontemporal_*; HIP float4 is a
// struct). Lowers to b128 loads/stores.
typedef float v4f __attribute__((ext_vector_type(4)));

__device__ __forceinline__ int clampi(int v, int lo, int hi) {
    return v < lo ? lo : (v > hi ? hi : v);
}

#if defined(__gfx1250__)
// LDS byte offset of a pointer into a __shared__ array (addrspace(3) is
// 32-bit; the cast yields the workgroup-relative LDS byte address).
__device__ __forceinline__ unsigned lds_byte_addr(const void* p) {
    return (unsigned)(unsigned long long)
        (__attribute__((address_space(3))) const void*)p;
}
#endif

// acc[c] += w * cell[c], 12 channels, channel-contiguous & 16B aligned in LDS
// -> lowers to 3x ds_load_b128 + 12 FMAs.
__device__ __forceinline__ void accum12(const float* cell, float w, float a[12]) {
    const v4f* c4 = (const v4f*)cell;
    v4f u = c4[0], v = c4[1], q = c4[2];
    a[0] += w * u.x;  a[1] += w * u.y;  a[2]  += w * u.z;  a[3]  += w * u.w;
    a[4] += w * v.x;  a[5] += w * v.y;  a[6]  += w * v.z;  a[7]  += w * v.w;
    a[8] += w * q.x;  a[9] += w * q.y;  a[10] += w * q.z;  a[11] += w * q.w;
}

__global__ __launch_bounds__(TPB)
void slice_apply_kernel(const float* __restrict__ grid,   // [B,12,8,16,16]
                        const float* __restrict__ guide,  // [B,1,1024,1024]
                        const float* __restrict__ fr,     // [B,3,1024,1024]
                        float* __restrict__ out)          // [B,3,1024,1024]
{
    // LDS copy of this batch's grid: [z][y][x][c], z-stride padded.
    __shared__ __align__(16) float s_grid[LDS_FLOATS];

    const int b   = blockIdx.y;
    const int tid = threadIdx.x;
    const float* gbase = grid + (size_t)b * GRID_PER_BATCH;

    // ---- Stage grid into LDS with async global->LDS DMA, transposing. ----
    // Source linear index t = ((c*8 + z)*16 + y)*16 + x  (coalesced per wave).
#if defined(__gfx1250__)
    const unsigned long long gbase_u64 = (unsigned long long)(uintptr_t)gbase;
#endif
    for (int t = tid; t < GRID_PER_BATCH; t += TPB) {
        const int x = t & 15;
        const int y = (t >> 4) & 15;
        const int z = (t >> 8) & 7;
        const int c = t >> 11;
        const int lidx = z * ZSTRIDE + ((y * GW + x) * GC + c);
#if defined(__gfx1250__)
        const unsigned laddr = lds_byte_addr(&s_grid[lidx]);
        const unsigned goff  = (unsigned)t * 4u;
        // vdst = LDS byte addr, vaddr = 32-bit byte offset, saddr = 64-bit base
        asm volatile("global_load_async_to_lds_b32 %0, %1, %2"
                     :
                     : "v"(laddr), "v"(goff), "s"(gbase_u64)
                     : "memory");
#else
        s_grid[lidx] = gbase[t];
#endif
    }
#if defined(__gfx1250__)
    asm volatile("s_wait_asynccnt 0" ::: "memory");
#endif
    __syncthreads();

    // ---- Streaming phase: each thread handles SWEEPS quads of 4 pixels. ----
    const float* guide_b = guide + (size_t)b * IMG_HW;
    const float* fr_b    = fr    + (size_t)b * 3 * IMG_HW;
    float*       out_b   = out   + (size_t)b * 3 * IMG_HW;

    const int pix0 = blockIdx.x * PIX_PER_BLOCK;

#pragma unroll
    for (int s = 0; s < SWEEPS; ++s) {
        const int pix = pix0 + s * (TPB * 4) + tid * 4;   // 4-aligned
        const int row = pix >> 10;                        // IMG_W == 1024
        const int col = pix & (IMG_W - 1);

        // Prefetch next sweep's streaming data (global_prefetch_b8).
        if (s + 1 < SWEEPS) {
            const int np = pix + TPB * 4;
            __builtin_prefetch(guide_b + np, 0, 0);
            __builtin_prefetch(fr_b + np, 0, 0);
            __builtin_prefetch(fr_b + IMG_HW + np, 0, 0);
            __builtin_prefetch(fr_b + 2 * IMG_HW + np, 0, 0);
        }

        // y interpolation (common per row)
        const float gy  = ((float)row + 0.5f) * ((float)GH / (float)IMG_H) - 0.5f;
        const float fy0 = floorf(gy);
        const float wy  = gy - fy0;
        const int   y0  = clampi((int)fy0,     0, GH - 1);
        const int   y1  = clampi((int)fy0 + 1, 0, GH - 1);

        // x cell (common to all 4 pixels of an aligned quad: cell-x boundaries
        // fall only at col % 32 == 0)
        const float gx0 = ((float)col + 0.5f) * ((float)GW / (float)IMG_W) - 0.5f;
        const float fx0 = floorf(gx0);
        const int   x0  = clampi((int)fx0,     0, GW - 1);
        const int   x1  = clampi((int)fx0 + 1, 0, GW - 1);

        // LDS float-index bases of the 4 xy corner cells within a z-slice
        const int c00 = (y0 * GW + x0) * GC;
        const int c01 = (y0 * GW + x1) * GC;
        const int c10 = (y1 * GW + x0) * GC;
        const int c11 = (y1 * GW + x1) * GC;

        // Non-temporal streaming loads (TH_LOAD_NT): read-once data.
        const v4f gd = __builtin_nontemporal_load((const v4f*)(guide_b + pix));
        const v4f f0 = __builtin_nontemporal_load((const v4f*)(fr_b + pix));
        const v4f f1 = __builtin_nontemporal_load((const v4f*)(fr_b + IMG_HW + pix));
        const v4f f2 = __builtin_nontemporal_load((const v4f*)(fr_b + 2 * IMG_HW + pix));

        const float gvals[4] = {gd.x, gd.y, gd.z, gd.w};
        const float rv[4]    = {f0.x, f0.y, f0.z, f0.w};
        const float gv[4]    = {f1.x, f1.y, f1.z, f1.w};
        const float bv[4]    = {f2.x, f2.y, f2.z, f2.w};
        float o0[4], o1[4], o2[4];

#pragma unroll
        for (int p = 0; p < 4; ++p) {
            // exact per-pixel wx (matches reference arithmetic)
            const float gx = ((float)(col + p) + 0.5f) *
                             ((float)GW / (float)IMG_W) - 0.5f;
            const float wx = gx - fx0;

            const float gz  = gvals[p] * (float)GD - 0.5f;
            const float fz0 = floorf(gz);
            const float wz  = gz - fz0;
            const int   z0  = clampi((int)fz0,     0, GD - 1);
            const int   z1  = clampi((int)fz0 + 1, 0, GD - 1);
            const int   zb0 = z0 * ZSTRIDE;
            const int   zb1 = z1 * ZSTRIDE;

            const float w00 = (1.0f - wy) * (1.0f - wx);
            const float w01 = (1.0f - wy) * wx;
            const float w10 = wy * (1.0f - wx);
            const float w11 = wy * wx;
            const float wzc0 = 1.0f - wz;

            float a[12];
#pragma unroll
            for (int c = 0; c < 12; ++c) a[c] = 0.0f;

            accum12(&s_grid[zb0 + c00], wzc0 * w00, a);
            accum12(&s_grid[zb0 + c01], wzc0 * w01, a);
            accum12(&s_grid[zb0 + c10], wzc0 * w10, a);
            accum12(&s_grid[zb0 + c11], wzc0 * w11, a);
            accum12(&s_grid[zb1 + c00], wz * w00, a);
            accum12(&s_grid[zb1 + c01], wz * w01, a);
            accum12(&s_grid[zb1 + c10], wz * w10, a);
            accum12(&s_grid[zb1 + c11], wz * w11, a);

            // out[o] = sum_i co[o*4+i]*fr[i] + co[o*4+3]
            o0[p] = a[0] * rv[p] + a[1] * gv[p] + a[2]  * bv[p] + a[3];
            o1[p] = a[4] * rv[p] + a[5] * gv[p] + a[6]  * bv[p] + a[7];
            o2[p] = a[8] * rv[p] + a[9] * gv[p] + a[10] * bv[p] + a[11];
        }

        // Non-temporal streaming stores (TH_STORE_NT): write-once data.
        const v4f s0 = {o0[0], o0[1], o0[2], o0[3]};
        const v4f s1 = {o1[0], o1[1], o1[2], o1[3]};
        const v4f s2 = {o2[0], o2[1], o2[2], o2[3]};
        __builtin_nontemporal_store(s0, (v4f*)(out_b + pix));
        __builtin_nontemporal_store(s1, (v4f*)(out_b + IMG_HW + pix));
        __builtin_nontemporal_store(s2, (v4f*)(out_b + 2 * IMG_HW + pix));
    }
}

extern "C" void kernel_launch(void* const* d_in, const int* in_sizes, int n_in,
                              void* d_out, int out_size, void* d_ws, size_t ws_size,
                              hipStream_t stream) {
    const float* grid  = (const float*)d_in[0];  // [B,12,8,16,16]
    const float* guide = (const float*)d_in[1];  // [B,1,1024,1024]
    const float* fr    = (const float*)d_in[2];  // [B,3,1024,1024]
    float*       out   = (float*)d_out;          // [B,3,1024,1024]

    const int B = in_sizes[0] / GRID_PER_BATCH;  // = 4

    dim3 g(BLOCKS_PER_BATCH, (unsigned)B, 1);
    dim3 blk(TPB, 1, 1);
    slice_apply_kernel<<<g, blk, 0, stream>>>(grid, guide, fr, out);
}